// SubsetOperator_28286654611518
// MI455X (gfx1250) — compile-verified
//
#include <hip/hip_runtime.h>

// Soft top-K (Gumbel subset) relaxation for MI455X / gfx1250.
//
// Reference: s = scores + g; repeat K=16 times:
//   s += log(max(1 - onehot, tiny)); onehot = softmax(s / TAU); khot += onehot
// TAU == 1.  Output = khot, f32 [4096, 8192].
//
// Optimization: work in w = exp(s - m0).  softmax is invariant to positive
// scaling of w, and s += log(mask) <=> w *= mask.  No per-iteration rescale is
// needed: unselected elements keep mask ~= 1, so max(w) at round t equals
// exp(s_(t) - s_(1)) -- the spread of the top-16 order statistics (a few
// units) -- nowhere near f32 underflow.  Elements >87 below the row max flush
// at init, exactly as their exp() terms flush inside the reference softmax.
//
// This removes ALL transcendentals from the K-loop (one v_exp_f32 per element
// total) and keeps each 32 KB row resident in VGPRs for all 16 rounds: one
// read pass + one write pass => ~384 MB of HBM traffic, ~16.5 us floor at
// 23.3 TB/s.
//
// Per-thread partial sums are reduced as a balanced tree (depth 5, dual-issue
// friendly) instead of a serial 31-deep add chain; the wave-level reduction
// then runs on the matrix pipe via V_WMMA_F32_16X16X4_F32 (A = lane partials
// in K0/K2, B = ones => D row i = v[i] + v[i+16]); each lane sums its 8
// accumulator VGPRs and one ds_swizzle SWAPX16 combines the two 16-lane
// halves.  XDL ops co-execute with VALU, keeping the reduction off the
// critical path.  EXEC is all-ones at every WMMA (uniform control flow, full
// 256-thread blocks).

#define N_ROWS 4096
#define N_COLS 8192
#define K_SEL  16
#define TPB    256                 // 8 waves of 32
#define NWAVE  (TPB / 32)
#define EPT    (N_COLS / TPB)      // 32 elements per thread, kept in VGPRs
#define NV4    (EPT / 4)           // 8 float4 per thread
#define EPS_TINY 1.1754943508222875e-38f   // np.finfo(np.float32).tiny

typedef float v2f __attribute__((ext_vector_type(2)));
typedef float v8f __attribute__((ext_vector_type(8)));

#if __has_builtin(__builtin_amdgcn_wmma_f32_16x16x4_f32)
#define HAVE_WMMA_F32X4 1
#endif

// v + (v from lane^16): one DS-pipe swizzle, no index VGPR, no LDS memory.
__device__ __forceinline__ float xor16_add(float v) {
#if __has_builtin(__builtin_amdgcn_ds_swizzle)
  // group-of-32 mode: and_mask=0x1f, or_mask=0, xor_mask=0x10 (SWAPX16)
  int o = __builtin_amdgcn_ds_swizzle(__float_as_int(v), 0x401F);
  return v + __int_as_float(o);
#else
  return v + __shfl_xor(v, 16, 32);
#endif
}

// Full 32-lane sum, result broadcast to every lane.
__device__ __forceinline__ float wave_sum(float v) {
#ifdef HAVE_WMMA_F32X4
  v2f a; a[0] = v;    a[1] = 0.0f;   // A 16x4: K0 (lanes 0-15) / K2 (lanes 16-31) = v
  v2f b; b[0] = 1.0f; b[1] = 1.0f;   // B 4x16: all ones
  v8f c = {};
  // D[i][j] = sum_k A[i][k] = v[i] + v[i+16], broadcast over columns.
  c = __builtin_amdgcn_wmma_f32_16x16x4_f32(
      /*neg_a=*/false, a, /*neg_b=*/false, b,
      /*c_mod=*/(short)0, c, /*reuse_a=*/false, /*reuse_b=*/false);
  // Lanes 0-15 hold row-sums 0-7 in c[0..7]; lanes 16-31 hold rows 8-15.
  float r = ((c[0] + c[1]) + (c[2] + c[3])) + ((c[4] + c[5]) + (c[6] + c[7]));
  return xor16_add(r);               // combine the two half-row partial sums
#else
  #pragma unroll
  for (int off = 16; off > 0; off >>= 1) v += __shfl_xor(v, off, 32);
  return v;
#endif
}

// Full 32-lane max, result broadcast to every lane (used once, at init).
__device__ __forceinline__ float wave_max(float v) {
  #pragma unroll
  for (int off = 16; off > 0; off >>= 1) v = fmaxf(v, __shfl_xor(v, off, 32));
  return v;
}

__global__ void __launch_bounds__(TPB, 1)
subset_topk_khot(const float* __restrict__ scores,
                 const float* __restrict__ gnoise,
                 float* __restrict__ out) {
  // Parity-double-buffered cross-wave reduction slots: one s_barrier per
  // iteration (iteration t writes slot t&1; the barrier at t+1 orders reads
  // of slot p against the rewrite at t+2).
  __shared__ float red[2][NWAVE];

  const int tid  = threadIdx.x;
  const int wave = tid >> 5;
  const int lane = tid & 31;

  const float4* s4 = reinterpret_cast<const float4*>(scores) + (size_t)blockIdx.x * (N_COLS / 4);
  const float4* g4 = reinterpret_cast<const float4*>(gnoise) + (size_t)blockIdx.x * (N_COLS / 4);
  float4*       o4 = reinterpret_cast<float4*>(out)          + (size_t)blockIdx.x * (N_COLS / 4);

  float w[EPT];    // holds s0 during load, then w = exp(s0 - m0)
  float kh[EPT];   // khot accumulator

  // ---- Load row (coalesced float4, stride TPB), s0 = scores + g, local max.
  float lmx = -3.402823466e38f;
  #pragma unroll
  for (int i = 0; i < NV4; ++i) {
    float4 a = s4[tid + i * TPB];
    float4 b = g4[tid + i * TPB];
    float x0 = a.x + b.x, x1 = a.y + b.y, x2 = a.z + b.z, x3 = a.w + b.w;
    w[4 * i + 0] = x0; w[4 * i + 1] = x1; w[4 * i + 2] = x2; w[4 * i + 3] = x3;
    lmx = fmaxf(lmx, fmaxf(fmaxf(x0, x1), fmaxf(x2, x3)));
  }

  // ---- Block-wide max of s0 (overflow guard for the single exp pass).
  lmx = wave_max(lmx);
  if (lane == 0) red[0][wave] = lmx;
  __syncthreads();
  float m0 = red[0][0];
  #pragma unroll
  for (int i = 1; i < NWAVE; ++i) m0 = fmaxf(m0, red[0][i]);
  __syncthreads();   // protect red[0] before the K-loop's first write

  // ---- One exp per element for the entire kernel.
  #pragma unroll
  for (int j = 0; j < EPT; ++j) {
    w[j]  = __expf(w[j] - m0);
    kh[j] = 0.0f;
  }

  // ---- K selection rounds: no transcendentals, one block-sum each.
  #pragma unroll 1
  for (int it = 0; it < K_SEL; ++it) {
    // Balanced-tree local sum: depth 5, independent adds (dual-issue), vs. a
    // serial 31-deep dependence chain the compiler cannot reassociate.
    float q[NV4];
    #pragma unroll
    for (int i = 0; i < NV4; ++i)
      q[i] = (w[4 * i + 0] + w[4 * i + 1]) + (w[4 * i + 2] + w[4 * i + 3]);
    float lsum = ((q[0] + q[1]) + (q[2] + q[3])) + ((q[4] + q[5]) + (q[6] + q[7]));

    lsum = wave_sum(lsum);   // matrix-pipe reduction (v_wmma_f32_16x16x4_f32)

    const int p = it & 1;
    if (lane == 0) red[p][wave] = lsum;
    __syncthreads();
    float Z = red[p][0];
    #pragma unroll
    for (int i = 1; i < NWAVE; ++i) Z += red[p][i];

    const float invZ = 1.0f / Z;   // softmax normalizer
    #pragma unroll
    for (int j = 0; j < EPT; ++j) {
      float oh = w[j] * invZ;                    // onehot = softmax(s)
      kh[j]   += oh;                             // khot accumulation
      w[j]    *= fmaxf(1.0f - oh, EPS_TINY);     // w *= mask  (== s += log mask)
    }
  }

  // ---- Store khot (coalesced float4).
  #pragma unroll
  for (int i = 0; i < NV4; ++i) {
    float4 o;
    o.x = kh[4 * i + 0]; o.y = kh[4 * i + 1];
    o.z = kh[4 * i + 2]; o.w = kh[4 * i + 3];
    o4[tid + i * TPB] = o;
  }
}

extern "C" void kernel_launch(void* const* d_in, const int* in_sizes, int n_in,
                              void* d_out, int out_size, void* d_ws, size_t ws_size,
                              hipStream_t stream) {
  (void)in_sizes; (void)n_in; (void)out_size; (void)d_ws; (void)ws_size;
  const float* scores = (const float*)d_in[0];
  const float* gnoise = (const float*)d_in[1];
  float*       out    = (float*)d_out;

  hipLaunchKernelGGL(subset_topk_khot,
                     dim3(N_ROWS), dim3(TPB), 0, stream,
                     scores, gnoise, out);
}